// GRUAttnDecoder_90967407329864
// MI455X (gfx1250) — compile-verified
//
#include <hip/hip_runtime.h>
#include <math.h>

#define BB    16
#define ENC   512
#define TT    64
#define DIM   768
#define NVOC  21128
#define NTILES 1321            // ceil(21128/16)
#define PSLOTS 128             // vocab partial slots per row (16 chunks * 8 waves)
#define LDSSTR 772             // 768 + 4 pad -> conflict-free fragment reads
#define SCALE 0.03608439182435161f   // 1/sqrt(768)
#define NEGBIG (-1e30f)

typedef __attribute__((ext_vector_type(2))) float v2f;
typedef __attribute__((ext_vector_type(8))) float v8f;

__device__ __forceinline__ float sigm(float x){ return 1.0f/(1.0f+expf(-x)); }

__device__ __forceinline__ float bfly_sum32(float v){
  #pragma unroll
  for (int m=16;m>=1;m>>=1) v += __shfl_xor(v,m);
  return v;
}
__device__ __forceinline__ float bfly_max16(float v){
  #pragma unroll
  for (int m=8;m>=1;m>>=1) v = fmaxf(v,__shfl_xor(v,m));
  return v;
}
__device__ __forceinline__ float bfly_sum16(float v){
  #pragma unroll
  for (int m=8;m>=1;m>>=1) v += __shfl_xor(v,m);
  return v;
}

// ---------------- embedding gather: EMB[row] = emb_W[decode_input[row]] ----------
__global__ void k_gather(const int* __restrict__ ids, const float* __restrict__ embW,
                         float* __restrict__ EMB){
  int row = blockIdx.x;                       // 0..1023  (b*T + t)
  const float* src = embW + (size_t)ids[row]*DIM;
  float* dst = EMB + (size_t)row*DIM;
  for (int d = threadIdx.x; d < DIM; d += 256) dst[d] = src[d];
}

// ---------------- generic WMMA f32 GEMM:  C(1024xN) = A(1024x768) @ W[:,off:off+768].T + bias
// W row-major with row stride wstride; N = 768 (grid.y*8 waves = 48 n-tiles)
__global__ void __launch_bounds__(256) k_gemm_bias(
    const float* __restrict__ A, const float* __restrict__ W,
    const float* __restrict__ bias, float* __restrict__ C,
    int wstride, int wcoloff){
  __shared__ __align__(16) float Alds[16*LDSSTR];
  const int tid = threadIdx.x;
  const int rt  = blockIdx.x;                 // row tile 0..63
  const float* Arows = A + (size_t)rt*16*DIM;
  for (int i = tid; i < 16*DIM/4; i += 256){
    int r = (i*4)/DIM, c = (i*4)%DIM;
    *(float4*)(&Alds[r*LDSSTR + c]) = *(const float4*)(Arows + (size_t)r*DIM + c);
  }
  __syncthreads();
  const int wave = tid>>5, lane = tid&31;
  const int m = lane & 15, half = lane >> 4;
  const int ntile = blockIdx.y*8 + wave;      // 0..47
  const int nbase = ntile*16;
  const float* bp = W + (size_t)(nbase + m)*wstride + wcoloff + 2*half;
  const float* ap = &Alds[m*LDSSTR + 2*half];
  v8f acc = {0.f,0.f,0.f,0.f,0.f,0.f,0.f,0.f};
  #pragma unroll 4
  for (int k = 0; k < DIM; k += 4){
    v2f a  = *(const v2f*)(ap + k);
    v2f bb = *(const v2f*)(bp + k);
    acc = __builtin_amdgcn_wmma_f32_16x16x4_f32(false, a, false, bb,
                                                (short)0, acc, false, false);
  }
  const float bv = bias[nbase + m];
  #pragma unroll
  for (int v = 0; v < 8; v++){
    int mr = rt*16 + v + 8*half;
    C[(size_t)mr*DIM + nbase + m] = acc[v] + bv;
  }
}

// ---------------- sequential GRU+attention recurrence, one block per batch element
__global__ void __launch_bounds__(256) k_recur(
    const float* __restrict__ ctx, const float* __restrict__ cmask,
    const float* __restrict__ GEBr, const float* __restrict__ GEBu,
    const float* __restrict__ GEBc,
    const float* __restrict__ W_r, const float* __restrict__ W_u,
    const float* __restrict__ W_c,
    const float* __restrict__ init_h, float* __restrict__ H){
  __shared__ __align__(16) float hb0[DIM], hb1[DIM], av[DIM], rl[DIM];
  __shared__ float sc[ENC], red[256];
  const int b = blockIdx.x, tid = threadIdx.x;
  const int wave = tid>>5, lane = tid&31;
  for (int d = tid; d < DIM; d += 256) hb0[d] = init_h[d];
  __syncthreads();

  for (int t = 0; t < TT; t++){
    float* hc = (t & 1) ? hb1 : hb0;
    float* hn = (t & 1) ? hb0 : hb1;
    const int row = b*TT + t;

    // ---- attention scores over ENC
    for (int s = wave; s < ENC; s += 8){
      const float* cr = ctx + ((size_t)b*ENC + s)*DIM;
      float p = 0.f;
      for (int d = lane; d < DIM; d += 32) p += cr[d]*hc[d];
      p = bfly_sum32(p);
      if (lane == 0){
        float ma = (1.0f - cmask[b*ENC + s]) * NEGBIG;
        sc[s] = p*SCALE + ma;
      }
    }
    __syncthreads();
    // ---- softmax over ENC
    float lm = -3e38f;
    for (int s = tid; s < ENC; s += 256) lm = fmaxf(lm, sc[s]);
    red[tid] = lm; __syncthreads();
    for (int o = 128; o > 0; o >>= 1){ if (tid < o) red[tid] = fmaxf(red[tid], red[tid+o]); __syncthreads(); }
    const float MX = red[0]; __syncthreads();
    float ls = 0.f;
    for (int s = tid; s < ENC; s += 256){ float e = expf(sc[s]-MX); sc[s] = e; ls += e; }
    red[tid] = ls; __syncthreads();
    for (int o = 128; o > 0; o >>= 1){ if (tid < o) red[tid] += red[tid+o]; __syncthreads(); }
    const float SE = red[0]; __syncthreads();
    // ---- attn_vec
    for (int d = tid; d < DIM; d += 256){
      const float* cb = ctx + (size_t)b*ENC*DIM + d;
      float a = 0.f;
      #pragma unroll 4
      for (int s = 0; s < ENC; s++) a += sc[s]*cb[(size_t)s*DIM];
      av[d] = a/SE;
    }
    __syncthreads();

    // ---- gates r,u (attn + h contributions; emb part precomputed in GEB*)
    float uval[3];
    #pragma unroll
    for (int i = 0; i < 3; i++){
      const int n = tid + i*256;
      const float* wr = W_r + (size_t)n*3*DIM;
      const float* wu = W_u + (size_t)n*3*DIM;
      float ar = GEBr[(size_t)row*DIM + n];
      float au = GEBu[(size_t)row*DIM + n];
      for (int k = 0; k < DIM; k += 4){
        float4 a = *(const float4*)(av + k);
        float4 w1 = *(const float4*)(wr + k);
        float4 w2 = *(const float4*)(wu + k);
        ar += a.x*w1.x + a.y*w1.y + a.z*w1.z + a.w*w1.w;
        au += a.x*w2.x + a.y*w2.y + a.z*w2.z + a.w*w2.w;
      }
      for (int k = 0; k < DIM; k += 4){
        float4 hh = *(const float4*)(hc + k);
        float4 w1 = *(const float4*)(wr + 2*DIM + k);
        float4 w2 = *(const float4*)(wu + 2*DIM + k);
        ar += hh.x*w1.x + hh.y*w1.y + hh.z*w1.z + hh.w*w1.w;
        au += hh.x*w2.x + hh.y*w2.y + hh.z*w2.z + hh.w*w2.w;
      }
      rl[n] = sigm(ar);
      uval[i] = sigm(au);
    }
    __syncthreads();
    // ---- candidate c and h update
    #pragma unroll
    for (int i = 0; i < 3; i++){
      const int n = tid + i*256;
      const float* wc = W_c + (size_t)n*3*DIM;
      float ac = GEBc[(size_t)row*DIM + n];
      for (int k = 0; k < DIM; k += 4){
        float4 a = *(const float4*)(av + k);
        float4 w1 = *(const float4*)(wc + k);
        ac += a.x*w1.x + a.y*w1.y + a.z*w1.z + a.w*w1.w;
      }
      for (int k = 0; k < DIM; k += 4){
        float4 hh = *(const float4*)(hc + k);
        float4 rr = *(const float4*)(rl + k);
        float4 w1 = *(const float4*)(wc + 2*DIM + k);
        ac += hh.x*rr.x*w1.x + hh.y*rr.y*w1.y + hh.z*rr.z*w1.z + hh.w*rr.w*w1.w;
      }
      const float c = tanhf(ac);
      const float u = uval[i];
      const float hv = (1.0f - u)*hc[n] + u*c;
      hn[n] = hv;
      H[(size_t)row*DIM + n] = hv;
    }
    __syncthreads();
  }
}

// ---------------- mode gate: sigmoid(H @ W_mode.T + b_mode), one warp per row
__global__ void k_mode(const float* __restrict__ H, const float* __restrict__ wm,
                       const float* __restrict__ bm, float* __restrict__ MODE){
  const int gw = (blockIdx.x*blockDim.x + threadIdx.x) >> 5;
  const int lane = threadIdx.x & 31;
  if (gw >= BB*TT) return;
  const float* h = H + (size_t)gw*DIM;
  float p = 0.f;
  for (int d = lane; d < DIM; d += 32) p += h[d]*wm[d];
  p = bfly_sum32(p);
  if (lane == 0) MODE[gw] = sigm(p + bm[0]);
}

// ---------------- copy distribution: sum of input_prob over positions matching target
__global__ void __launch_bounds__(256) k_copy(
    const float* __restrict__ CPQ, const float* __restrict__ ctx,
    const float* __restrict__ cmask, const int* __restrict__ ids,
    const int* __restrict__ tgt, float* __restrict__ CPF){
  __shared__ float q[DIM]; __shared__ float sc[ENC]; __shared__ float red[256];
  const int b = blockIdx.x, t = blockIdx.y, tid = threadIdx.x;
  const int row = b*TT + t;
  for (int d = tid; d < DIM; d += 256) q[d] = CPQ[(size_t)row*DIM + d];
  __syncthreads();
  const int wave = tid>>5, lane = tid&31;
  for (int s = wave; s < ENC; s += 8){
    const float* cr = ctx + ((size_t)b*ENC + s)*DIM;
    float p = 0.f;
    for (int d = lane; d < DIM; d += 32) p += cr[d]*q[d];
    p = bfly_sum32(p);
    if (lane == 0) sc[s] = p + (1.0f - cmask[b*ENC + s])*NEGBIG;
  }
  __syncthreads();
  float lm = -3e38f;
  for (int s = tid; s < ENC; s += 256) lm = fmaxf(lm, sc[s]);
  red[tid] = lm; __syncthreads();
  for (int o = 128; o > 0; o >>= 1){ if (tid < o) red[tid] = fmaxf(red[tid], red[tid+o]); __syncthreads(); }
  const float MX = red[0]; __syncthreads();
  float ls = 0.f;
  for (int s = tid; s < ENC; s += 256){ float e = expf(sc[s]-MX); sc[s] = e; ls += e; }
  red[tid] = ls; __syncthreads();
  for (int o = 128; o > 0; o >>= 1){ if (tid < o) red[tid] += red[tid+o]; __syncthreads(); }
  const float SE = red[0]; __syncthreads();
  const int tg = tgt[row];
  float ln = 0.f;
  for (int s = tid; s < ENC; s += 256) if (ids[b*ENC + s] == tg) ln += sc[s];
  red[tid] = ln; __syncthreads();
  for (int o = 128; o > 0; o >>= 1){ if (tid < o) red[tid] += red[tid+o]; __syncthreads(); }
  if (tid == 0) CPF[row] = red[0]/SE;
}

// ---------------- vocab GEMM (WMMA f32) with online-softmax partials ----------
// M-blocked: 32 A-rows (2 row-tiles) staged in LDS; each wave keeps two
// accumulators sharing one global B fragment -> halves L2 B traffic.
__global__ void __launch_bounds__(256) k_vocab(
    const float* __restrict__ OUT, const float* __restrict__ embW,
    float* __restrict__ part){
  __shared__ __align__(16) float Alds[32*LDSSTR];   // 98.8 KB (CDNA5: 320KB/WGP)
  const int tid = threadIdx.x;
  const int rt  = blockIdx.x;                 // row-pair tile 0..31 (rows rt*32..rt*32+31)
  const float* Arows = OUT + (size_t)rt*32*DIM;
  for (int i = tid; i < 32*DIM/4; i += 256){
    int r = (i*4)/DIM, c = (i*4)%DIM;
    *(float4*)(&Alds[r*LDSSTR + c]) = *(const float4*)(Arows + (size_t)r*DIM + c);
  }
  __syncthreads();
  const int wave = tid>>5, lane = tid&31;
  const int m = lane & 15, half = lane >> 4;
  const int slot = blockIdx.y*8 + wave;       // 0..PSLOTS-1
  float mx0[8], se0[8], mx1[8], se1[8];
  #pragma unroll
  for (int v = 0; v < 8; v++){ mx0[v] = -3e38f; se0[v] = 0.f; mx1[v] = -3e38f; se1[v] = 0.f; }
  const float* ap0 = &Alds[m*LDSSTR + 2*half];
  const float* ap1 = ap0 + 16*LDSSTR;

  for (int tile = slot; tile < NTILES; tile += PSLOTS){
    const int nbase = tile*16;
    const int n = nbase + m;
    const bool valid = (n < NVOC);
    const float* bp = embW + (size_t)(valid ? n : (NVOC-1))*DIM + 2*half;
    v8f acc0 = {0.f,0.f,0.f,0.f,0.f,0.f,0.f,0.f};
    v8f acc1 = {0.f,0.f,0.f,0.f,0.f,0.f,0.f,0.f};
    #pragma unroll 4
    for (int k = 0; k < DIM; k += 4){
      v2f bb = *(const v2f*)(bp + k);     // one B fragment feeds two WMMAs
      v2f a0 = *(const v2f*)(ap0 + k);
      v2f a1 = *(const v2f*)(ap1 + k);
      acc0 = __builtin_amdgcn_wmma_f32_16x16x4_f32(false, a0, false, bb,
                                                   (short)0, acc0, false, false);
      acc1 = __builtin_amdgcn_wmma_f32_16x16x4_f32(false, a1, false, bb,
                                                   (short)0, acc1, false, false);
    }
    // online softmax merge per output row (lane's column n = nbase + m)
    #pragma unroll
    for (int v = 0; v < 8; v++){
      float val0 = valid ? acc0[v] : -3e38f;
      float tm0 = bfly_max16(val0);
      float ts0 = bfly_sum16(expf(val0 - tm0));
      float nm0 = fmaxf(mx0[v], tm0);
      se0[v] = se0[v]*expf(mx0[v]-nm0) + ts0*expf(tm0-nm0);
      mx0[v] = nm0;

      float val1 = valid ? acc1[v] : -3e38f;
      float tm1 = bfly_max16(val1);
      float ts1 = bfly_sum16(expf(val1 - tm1));
      float nm1 = fmaxf(mx1[v], tm1);
      se1[v] = se1[v]*expf(mx1[v]-nm1) + ts1*expf(tm1-nm1);
      mx1[v] = nm1;
    }
  }
  if (m == 0){
    #pragma unroll
    for (int v = 0; v < 8; v++){
      const int row0 = rt*32 + v + 8*half;
      const int row1 = row0 + 16;
      size_t i0 = ((size_t)row0*PSLOTS + slot)*2;
      size_t i1 = ((size_t)row1*PSLOTS + slot)*2;
      part[i0] = mx0[v]; part[i0+1] = se0[v];
      part[i1] = mx1[v]; part[i1+1] = se1[v];
    }
  }
}

// ---------------- combine partials, compute per-row NLL, accumulate ----------
__global__ void k_loss(const float* __restrict__ part, const float* __restrict__ OUT,
                       const float* __restrict__ embW, const float* __restrict__ MODE,
                       const float* __restrict__ CPF, const int* __restrict__ tgt,
                       float* __restrict__ acc2){
  const int gw = (blockIdx.x*blockDim.x + threadIdx.x) >> 5;
  const int lane = threadIdx.x & 31;
  const int nwarp = (gridDim.x*blockDim.x) >> 5;
  for (int row = gw; row < BB*TT; row += nwarp){
    float lm = -3e38f;
    for (int i = lane; i < PSLOTS; i += 32)
      lm = fmaxf(lm, part[((size_t)row*PSLOTS + i)*2]);
    #pragma unroll
    for (int msk = 16; msk >= 1; msk >>= 1) lm = fmaxf(lm, __shfl_xor(lm, msk));
    float ls = 0.f;
    for (int i = lane; i < PSLOTS; i += 32){
      float m_ = part[((size_t)row*PSLOTS + i)*2];
      float s_ = part[((size_t)row*PSLOTS + i)*2 + 1];
      ls += s_*expf(m_ - lm);
    }
    ls = bfly_sum32(ls);
    const int tg = tgt[row];
    const float* o = OUT + (size_t)row*DIM;
    const float* e = embW + (size_t)tg*DIM;
    float lt = 0.f;
    for (int d = lane; d < DIM; d += 32) lt += o[d]*e[d];
    lt = bfly_sum32(lt);
    if (lane == 0 && tg != 0){
      const float md = MODE[row];
      const float p = md*expf(lt - lm)/ls + (1.0f - md)*CPF[row];
      atomicAdd(&acc2[0], -logf(p + 1e-6f));
      atomicAdd(&acc2[1], 1.0f);
    }
  }
}

__global__ void k_final(const float* __restrict__ acc2, float* __restrict__ out){
  if (threadIdx.x < BB) out[threadIdx.x] = acc2[0]/acc2[1];
}

extern "C" void kernel_launch(void* const* d_in, const int* in_sizes, int n_in,
                              void* d_out, int out_size, void* d_ws, size_t ws_size,
                              hipStream_t stream) {
  (void)in_sizes; (void)n_in; (void)out_size; (void)ws_size;
  const int*   input_ids = (const int*)  d_in[0];
  const float* ctx       = (const float*)d_in[1];
  const float* cmask     = (const float*)d_in[2];
  const int*   dec_in    = (const int*)  d_in[3];
  const int*   dec_tgt   = (const int*)  d_in[4];
  const float* embW      = (const float*)d_in[7];
  const float* W_r  = (const float*)d_in[8];  const float* b_r  = (const float*)d_in[9];
  const float* W_u  = (const float*)d_in[10]; const float* b_u  = (const float*)d_in[11];
  const float* W_c  = (const float*)d_in[12]; const float* b_c  = (const float*)d_in[13];
  const float* W_o  = (const float*)d_in[14]; const float* b_o  = (const float*)d_in[15];
  const float* W_cp = (const float*)d_in[16]; const float* b_cp = (const float*)d_in[17];
  const float* W_md = (const float*)d_in[18]; const float* b_md = (const float*)d_in[19];
  const float* init_h = (const float*)d_in[20];
  float* outp = (float*)d_out;

  const size_t R = (size_t)BB*TT;         // 1024 rows
  float* ws   = (float*)d_ws;
  float* EMB  = ws;              // 1024*768
  float* GEBr = EMB  + R*DIM;
  float* GEBu = GEBr + R*DIM;
  float* GEBc = GEBu + R*DIM;
  float* H    = GEBc + R*DIM;
  float* OUT  = H    + R*DIM;
  float* CPQ  = OUT  + R*DIM;
  float* MODE = CPQ  + R*DIM;    // 1024
  float* CPF  = MODE + R;        // 1024
  float* PART = CPF  + R;        // 1024*128*2
  float* ACC  = PART + R*PSLOTS*2;  // 2

  // 1. gather decode embeddings
  k_gather<<<dim3(BB*TT), dim3(256), 0, stream>>>(dec_in, embW, EMB);
  // 2. precompute embedding contribution of each gate (WMMA), bias folded in
  k_gemm_bias<<<dim3(64,6), dim3(256), 0, stream>>>(EMB, W_r, b_r, GEBr, 3*DIM, DIM);
  k_gemm_bias<<<dim3(64,6), dim3(256), 0, stream>>>(EMB, W_u, b_u, GEBu, 3*DIM, DIM);
  k_gemm_bias<<<dim3(64,6), dim3(256), 0, stream>>>(EMB, W_c, b_c, GEBc, 3*DIM, DIM);
  // 3. sequential recurrence (one block per batch element)
  k_recur<<<dim3(BB), dim3(256), 0, stream>>>(ctx, cmask, GEBr, GEBu, GEBc,
                                              W_r, W_u, W_c, init_h, H);
  // 4. output / copy-query projections (WMMA)
  k_gemm_bias<<<dim3(64,6), dim3(256), 0, stream>>>(H, W_o,  b_o,  OUT, DIM, 0);
  k_gemm_bias<<<dim3(64,6), dim3(256), 0, stream>>>(H, W_cp, b_cp, CPQ, DIM, 0);
  // 5. mode gate
  k_mode<<<dim3(128), dim3(256), 0, stream>>>(H, W_md, b_md, MODE);
  // 6. copy distribution mass on target token
  k_copy<<<dim3(BB,TT), dim3(256), 0, stream>>>(CPQ, ctx, cmask, input_ids, dec_tgt, CPF);
  // 7. big vocab GEMM with online softmax partials (WMMA, M-blocked)
  k_vocab<<<dim3(32,PSLOTS/8), dim3(256), 0, stream>>>(OUT, embW, PART);
  // 8. loss accumulation
  hipMemsetAsync(ACC, 0, 2*sizeof(float), stream);
  k_loss<<<dim3(32), dim3(256), 0, stream>>>(PART, OUT, embW, MODE, CPF, dec_tgt, ACC);
  k_final<<<dim3(1), dim3(32), 0, stream>>>(ACC, outp);
}